// BoxRenderLoss_55276229099895
// MI455X (gfx1250) — compile-verified
//
#include <hip/hip_runtime.h>

// BoxRenderLoss fused kernel for gfx1250 (MI455X, wave32, WMMA).
// dist(f,p) = |f|^2 + |p|^2 - 2 f.p ; the (-2 f.p + |p|^2) term is computed
// with V_WMMA_F32_16X16X4_F32 (A = fragments 16x4 with K=0,1 used,
// B = -2*points 4x16, C = |p|^2 broadcast). One wave = one (box, direction).

#define FPN 100   // fragment samples per box (10x10)
#define BPN 100   // boundary samples per box

typedef __attribute__((ext_vector_type(2))) float v2f;
typedef __attribute__((ext_vector_type(8))) float v8f;

__device__ __forceinline__ float2 frag_coord(int m) {
  // meshgrid(linspace(0,1,10), linspace(0,1,10), 'ij') flattened: (i/9, j/9)
  int i = m / 10;
  int j = m - i * 10;
  return make_float2((float)i * (1.0f / 9.0f), (float)j * (1.0f / 9.0f));
}

__device__ __forceinline__ float2 bnd_coord(int n) {
  // first 50: (0, j/24) then (1, j/24); last 50: (i/24, 0)/(i/24, 1) interleaved
  float x, y;
  if (n < 50) {
    int j = (n < 25) ? n : (n - 25);
    x = (n < 25) ? 0.0f : 1.0f;
    y = (float)j * (1.0f / 24.0f);
  } else {
    int k = n - 50;
    x = (float)(k >> 1) * (1.0f / 24.0f);
    y = (k & 1) ? 1.0f : 0.0f;
  }
  return make_float2(x, y);
}

__global__ void box_render_loss_init(float* out, int n) {
  int i = blockIdx.x * blockDim.x + threadIdx.x;
  if (i < n) out[i] = 0.0f;
}

__global__ __launch_bounds__(256) void box_render_loss_wmma(
    const float* __restrict__ boxes, const float* __restrict__ targets,
    float* __restrict__ out, int B, float scale) {
  const int tid  = blockIdx.x * blockDim.x + threadIdx.x;
  const int wave = tid >> 5;
  const int lane = tid & 31;
  const int dir  = wave & 1;
  const int bidx = wave >> 1;
  if (bidx >= B) return;  // wave-uniform exit; survivors keep EXEC all-ones

  const float4 bx = reinterpret_cast<const float4*>(boxes)[bidx];
  const float4 tg = reinterpret_cast<const float4*>(targets)[bidx];
  const float4 src = dir ? tg : bx;  // fragments sampled inside this box
  const float4 dst = dir ? bx : tg;  // boundary points + outside-mask box

  const float s_lox = src.x, s_loy = src.y;
  const float s_whx = src.z - src.x, s_why = src.w - src.y;
  const float d_lox = dst.x, d_loy = dst.y;
  const float d_whx = dst.z - dst.x, d_why = dst.w - dst.y;

  const int half   = (lane >> 4) & 1;
  const int lane16 = lane & 15;

  float acc = 0.0f;

  for (int Mt = 0; Mt < 7; ++Mt) {
    // ---- A matrix (16x4 f32): lanes 0-15 hold K=0,1 = (fx,fy); lanes 16-31 K=2,3 = 0
    v2f a; a.x = 0.0f; a.y = 0.0f;
    {
      int m = Mt * 16 + lane;
      if (lane < 16 && m < FPN) {
        float2 f = frag_coord(m);
        a.x = f.x * s_whx + s_lox;
        a.y = f.y * s_why + s_loy;
      }
    }
    // ---- per-row |f|^2 and outside-mask (C/D row r <-> M = r + 8*half)
    float fmag[8], outm[8];
#pragma unroll
    for (int r = 0; r < 8; ++r) {
      int m = Mt * 16 + r + half * 8;
      fmag[r] = 0.0f;
      outm[r] = 0.0f;
      if (m < FPN) {
        float2 f = frag_coord(m);
        float fx = f.x * s_whx + s_lox;
        float fy = f.y * s_why + s_loy;
        fmag[r] = fx * fx + fy * fy;
        bool inside = (fx - dst.x >= 0.0f) && (fy - dst.y >= 0.0f) &&
                      (dst.z - fx >= 0.0f) && (dst.w - fy >= 0.0f);
        outm[r] = inside ? 0.0f : 1.0f;
      }
    }

    float minv[8];
#pragma unroll
    for (int r = 0; r < 8; ++r) minv[r] = 3.0e38f;

    for (int Nt = 0; Nt < 7; ++Nt) {
      // ---- B matrix (4x16 f32): rows K=0,1 = (-2*px, -2*py); K=2,3 zero
      v2f b; b.x = 0.0f; b.y = 0.0f;
      {
        int n = Nt * 16 + lane;
        if (lane < 16 && n < BPN) {
          float2 p = bnd_coord(n);
          float px = p.x * d_whx + d_lox;
          float py = p.y * d_why + d_loy;
          b.x = -2.0f * px;
          b.y = -2.0f * py;
        }
      }
      // ---- C matrix: |p|^2 broadcast down all 8 rows (per-lane column N)
      float pm = 1.0e30f;  // padded columns can never win the min
      {
        int n = Nt * 16 + lane16;
        if (n < BPN) {
          float2 p = bnd_coord(n);
          float px = p.x * d_whx + d_lox;
          float py = p.y * d_why + d_loy;
          pm = px * px + py * py;
        }
      }
      v8f c;
#pragma unroll
      for (int r = 0; r < 8; ++r) c[r] = pm;

      // D = A*B + C = |p|^2 - 2 f.p per (M,N) tile element
      v8f d = __builtin_amdgcn_wmma_f32_16x16x4_f32(
          false, a, false, b, (short)0, c, false, false);
#pragma unroll
      for (int r = 0; r < 8; ++r) minv[r] = fminf(minv[r], d[r]);
    }

    // full min over all N: butterfly min across the 16 lanes of each half
#pragma unroll
    for (int off = 1; off < 16; off <<= 1) {
#pragma unroll
      for (int r = 0; r < 8; ++r)
        minv[r] = fminf(minv[r], __shfl_xor(minv[r], off, 32));
    }
#pragma unroll
    for (int r = 0; r < 8; ++r)
      acc += (fmag[r] + minv[r]) * outm[r];
  }

  // each half's 16 lanes are identical -> wave-sum over-counts 16x (folded into scale)
#pragma unroll
  for (int off = 16; off >= 1; off >>= 1)
    acc += __shfl_xor(acc, off, 32);
  if (lane == 0) atomicAdd(out, acc * scale);
}

extern "C" void kernel_launch(void* const* d_in, const int* in_sizes, int n_in,
                              void* d_out, int out_size, void* d_ws, size_t ws_size,
                              hipStream_t stream) {
  (void)n_in; (void)d_ws; (void)ws_size;
  const float* boxes   = (const float*)d_in[0];
  const float* targets = (const float*)d_in[1];
  float* out = (float*)d_out;
  const int B = in_sizes[0] / 4;  // 4096

  box_render_loss_init<<<1, 256, 0, stream>>>(out, out_size);

  const int waves   = 2 * B;          // (box, direction) per wave
  const int threads = waves * 32;
  const int block   = 256;            // 8 wave32 per block
  const int grid    = (threads + block - 1) / block;
  // 1/(2*B*FP) for the reference normalization, extra /16 for duplicated lanes
  const float scale = 1.0f / (2.0f * (float)B * (float)FPN * 16.0f);

  box_render_loss_wmma<<<grid, block, 0, stream>>>(boxes, targets, out, B, scale);
}